// TopkQuantLayer_53635551592630
// MI455X (gfx1250) — compile-verified
//
#include <hip/hip_runtime.h>
#include <stdint.h>

// ---------------------------------------------------------------------------
// Sorted-chunk 4-bit quantization without sorting (exact order statistics).
//   mn_k = sorted[k*C], mx_k = sorted[(k+1)*C-1]; chunk(x) = #{k: mn_k <= x};
//   q = rint((x-mn)/step)*step + mn.
// Traffic: 3 streaming reads + 1 write of 256MB ~ 1.28GB @ 23.3TB/s ~ 55us.
// CDNA5 paths: 256KB LDS histograms (320KB/WGP), TDM tensor_load_to_lds with
// s_wait_tensorcnt double-buffering, wave32, non-temporal stores.
// ---------------------------------------------------------------------------

#define HBINS 65536u
#define NCHUNK 16u
#define TILE 16384u  // floats per TDM tile (64KB)

typedef __attribute__((ext_vector_type(4))) uint32_t v4u;
typedef __attribute__((ext_vector_type(8))) uint32_t v8u;
typedef __attribute__((ext_vector_type(4))) float v4f;

__device__ __forceinline__ uint32_t f2key(uint32_t u) {
  uint32_t m = (u >> 31) ? 0xFFFFFFFFu : 0x80000000u;  // order-preserving map
  return u ^ m;
}
__device__ __forceinline__ uint32_t key2f(uint32_t k) {
  uint32_t m = (k >> 31) ? 0x80000000u : 0xFFFFFFFFu;
  return k ^ m;
}

// ------------------------------- init --------------------------------------
__global__ __launch_bounds__(256) void k_init(uint32_t* w, uint32_t nwords) {
  uint32_t stride = gridDim.x * blockDim.x;
  for (uint32_t i = blockIdx.x * blockDim.x + threadIdx.x; i < nwords; i += stride)
    w[i] = 0u;
}

// -------------------- pass 1: hi-16 histogram ------------------------------
__global__ __launch_bounds__(256) void k_hist_hi(const float* __restrict__ x,
                                                 uint32_t n,
                                                 uint32_t* __restrict__ histHi) {
  __shared__ uint32_t sh[HBINS];  // 256KB private histogram (CDNA5 320KB LDS)
  for (uint32_t i = threadIdx.x; i < HBINS; i += blockDim.x) sh[i] = 0u;
  __syncthreads();

  const uint4* x4 = (const uint4*)x;
  uint32_t n4 = n >> 2;
  uint32_t stride = gridDim.x * blockDim.x;
  uint32_t tid = blockIdx.x * blockDim.x + threadIdx.x;
  // 4x unrolled grid-stride: 4 B128 loads in flight per thread (MLP for HBM)
  for (uint32_t base = tid; base < n4; base += 4u * stride) {
    uint4 a[4];
    uint32_t cnt = 0;
#pragma unroll
    for (int u = 0; u < 4; ++u) {
      uint32_t idx = base + (uint32_t)u * stride;
      if (idx < n4) { a[cnt++] = x4[idx]; }
    }
    for (uint32_t u = 0; u < cnt; ++u) {
      atomicAdd(&sh[f2key(a[u].x) >> 16], 1u);
      atomicAdd(&sh[f2key(a[u].y) >> 16], 1u);
      atomicAdd(&sh[f2key(a[u].z) >> 16], 1u);
      atomicAdd(&sh[f2key(a[u].w) >> 16], 1u);
    }
  }
  __syncthreads();
  for (uint32_t i = threadIdx.x; i < HBINS; i += blockDim.x) {
    uint32_t c = sh[i];
    if (c) atomicAdd(&histHi[i], c);
  }
}

// ---------------- pass 2: locate boundary ranks in hi bins -----------------
// bounds[2*s] = hi bin of target s, bounds[2*s+1] = residual rank inside bin.
// targets s=0..15: rank k*C (mn_k); s=16..31: rank (k+1)*C-1 (mx_k).
__global__ __launch_bounds__(1024) void k_select_hi(const uint32_t* __restrict__ histHi,
                                                    uint32_t n,
                                                    uint32_t* __restrict__ bounds) {
  __shared__ uint32_t psum[1024];
  __shared__ uint32_t pref[1024];
  uint32_t t = threadIdx.x;
  uint32_t s = 0;
  for (int j = 0; j < 64; ++j) s += histHi[t * 64 + j];
  psum[t] = s;
  __syncthreads();
  if (t == 0) {
    uint32_t acc = 0;
    for (int i = 0; i < 1024; ++i) { pref[i] = acc; acc += psum[i]; }
  }
  __syncthreads();

  uint32_t C = n >> 4;
  uint32_t cum = pref[t];
  for (int j = 0; j < 64; ++j) {
    uint32_t b = t * 64 + j;
    uint32_t c = histHi[b];
    if (c) {
      for (uint32_t k = 0; k < NCHUNK; ++k) {
        uint32_t rmn = k * C;
        uint32_t rmx = (k + 1) * C - 1;
        if (rmn >= cum && rmn < cum + c) { bounds[2 * k] = b; bounds[2 * k + 1] = rmn - cum; }
        uint32_t sx = 16 + k;
        if (rmx >= cum && rmx < cum + c) { bounds[2 * sx] = b; bounds[2 * sx + 1] = rmx - cum; }
      }
    }
    cum += c;
  }
}

// -------- pass 3: lo-16 histograms restricted to the boundary bins ---------
__global__ __launch_bounds__(256) void k_hist_lo(const float* __restrict__ x,
                                                 uint32_t n,
                                                 const uint32_t* __restrict__ bounds,
                                                 uint32_t* __restrict__ histLo) {
  __shared__ uint32_t lut[HBINS];  // 256KB mask-per-hi-bin LUT
  for (uint32_t i = threadIdx.x; i < HBINS; i += blockDim.x) lut[i] = 0u;
  __syncthreads();
  if (threadIdx.x < 32) atomicOr(&lut[bounds[2 * threadIdx.x]], 1u << threadIdx.x);
  __syncthreads();

  const uint4* x4 = (const uint4*)x;
  uint32_t n4 = n >> 2;
  uint32_t stride = gridDim.x * blockDim.x;
  uint32_t tid = blockIdx.x * blockDim.x + threadIdx.x;
  for (uint32_t base = tid; base < n4; base += 4u * stride) {
    uint4 a[4];
    uint32_t cnt = 0;
#pragma unroll
    for (int u = 0; u < 4; ++u) {
      uint32_t idx = base + (uint32_t)u * stride;
      if (idx < n4) { a[cnt++] = x4[idx]; }
    }
    for (uint32_t u = 0; u < cnt; ++u) {
      uint32_t ks[4] = {f2key(a[u].x), f2key(a[u].y), f2key(a[u].z), f2key(a[u].w)};
#pragma unroll
      for (int e = 0; e < 4; ++e) {
        uint32_t m = lut[ks[e] >> 16];
        while (m) {  // rare: only boundary-bin elements (~1e-3 of data)
          uint32_t slot = (uint32_t)__builtin_ctz(m);
          atomicAdd(&histLo[slot * HBINS + (ks[e] & 0xFFFFu)], 1u);
          m &= m - 1u;
        }
      }
    }
  }
}

// ----------- pass 4: scan lo histograms -> exact boundary values -----------
__global__ __launch_bounds__(256) void k_select_lo(const uint32_t* __restrict__ bounds,
                                                   const uint32_t* __restrict__ histLo,
                                                   float* __restrict__ vals) {
  __shared__ uint32_t part[256];
  uint32_t slot = blockIdx.x;  // 0..31
  uint32_t bin = bounds[2 * slot];
  uint32_t resid = bounds[2 * slot + 1];
  const uint32_t* h = histLo + (size_t)slot * HBINS;
  uint32_t t = threadIdx.x;
  uint32_t s = 0;
  for (int j = 0; j < 256; ++j) s += h[t * 256 + j];
  part[t] = s;
  __syncthreads();
  if (t == 0) {
    uint32_t acc = 0, lo = 0xFFFFu;
    for (int i = 0; i < 256; ++i) {
      if (lo == 0xFFFFu && resid < acc + part[i]) {
        uint32_t c2 = acc;
        for (int j = 0; j < 256; ++j) {
          uint32_t c = h[i * 256 + j];
          if (resid < c2 + c) { lo = (uint32_t)(i * 256 + j); break; }
          c2 += c;
        }
      }
      acc += part[i];
    }
    uint32_t key = (bin << 16) | lo;
    vals[slot] = __uint_as_float(key2f(key));
  }
}

// --------------------------- build tables ----------------------------------
// tabs: [0..15]=mn, [16..31]=mx, [32..47]=safe step, [48..63]=chunk thresholds
__global__ void k_finalize(const float* __restrict__ vals, float* __restrict__ tabs) {
  uint32_t k = threadIdx.x;
  if (k < NCHUNK) {
    float mn = vals[k], mx = vals[16 + k];
    float st = (mx - mn) * (1.0f / 15.0f);
    tabs[k] = mn;
    tabs[16 + k] = mx;
    tabs[32 + k] = (st == 0.0f) ? 1.0f : st;
    tabs[48 + k] = mn;  // lower-bound threshold of chunk k
  }
}

// ----------------- TDM: issue tensor_load_to_lds of one tile ---------------
__device__ __forceinline__ void tdm_load_tile(const float* gsrc,
                                              uint32_t lds_off_bytes,
                                              uint32_t tile_elems) {
  uint64_t ga = (uint64_t)(uintptr_t)gsrc;
  v4u g0;
  g0[0] = 1u;                                 // count=1, user descriptor
  g0[1] = lds_off_bytes;                      // D#.lds_addr
  g0[2] = (uint32_t)ga;                       // global_addr[31:0]
  g0[3] = (uint32_t)(ga >> 32) | (2u << 30);  // global_addr[56:32] | type=2
  v8u g1;
  g1[0] = 0x00020000u;                        // data_size = 4 bytes
  g1[1] = (tile_elems & 0xFFFFu) << 16;       // tensor_dim0[15:0]
  g1[2] = (tile_elems >> 16) | (1u << 16);    // tensor_dim0[31:16] | tensor_dim1=1
  g1[3] = (tile_elems & 0xFFFFu) << 16;       // tile_dim0 (1-D tile)
  g1[4] = 0u;                                 // tile_dim1=0, tile_dim2=0
  g1[5] = tile_elems;                         // tensor_dim0_stride[31:0]
  g1[6] = 0u;
  g1[7] = 0u;
  asm volatile("tensor_load_to_lds %0, %1" ::"s"(g0), "s"(g1) : "memory");
}

// ---------------- pass 5: classify + quantize (TDM staged) -----------------
__global__ __launch_bounds__(256) void k_quant(const float* __restrict__ x,
                                               float* __restrict__ out,
                                               uint32_t n,
                                               const float* __restrict__ tabs) {
  __shared__ float smem[2 * TILE + 48];
  float* mnS = smem + 2 * TILE;
  float* mxS = mnS + 16;
  float* stS = mnS + 32;
  if (threadIdx.x < 48) mnS[threadIdx.x] = tabs[threadIdx.x];
  float thr[15];  // uniform -> registers
#pragma unroll
  for (int k = 0; k < 15; ++k) thr[k] = tabs[49 + k];
  __syncthreads();

  uint32_t ntiles = n / TILE;
  uint32_t per = (ntiles + gridDim.x - 1) / gridDim.x;
  uint32_t t0 = blockIdx.x * per;
  uint32_t t1 = (t0 + per < ntiles) ? (t0 + per) : ntiles;

  if (t0 < t1) {
    bool issuer = (threadIdx.x < 32);  // wave 0 only (wave-uniform branch)
    if (issuer)
      tdm_load_tile(x + (size_t)t0 * TILE, (uint32_t)(uintptr_t)smem, TILE);

    for (uint32_t t = t0; t < t1; ++t) {
      uint32_t bi = (t - t0) & 1u;
      float* cur = smem + bi * TILE;
      float* nxt = smem + (bi ^ 1u) * TILE;
      if (issuer) {
        if (t + 1 < t1) {
          tdm_load_tile(x + (size_t)(t + 1) * TILE, (uint32_t)(uintptr_t)nxt, TILE);
          __builtin_amdgcn_s_wait_tensorcnt(1);  // tile t done, t+1 in flight
        } else {
          __builtin_amdgcn_s_wait_tensorcnt(0);
        }
      }
      __syncthreads();

      const v4f* c4 = (const v4f*)cur;
      v4f* o4 = (v4f*)(out + (size_t)t * TILE);
      for (uint32_t i = threadIdx.x; i < TILE / 4; i += blockDim.x) {
        v4f v = c4[i];
        float r[4] = {v.x, v.y, v.z, v.w};
#pragma unroll
        for (int e = 0; e < 4; ++e) {
          float xv = r[e];
          int ch = 0;
#pragma unroll
          for (int k = 0; k < 15; ++k) ch += (xv >= thr[k]) ? 1 : 0;
          float mn = mnS[ch], mx = mxS[ch], st = stS[ch];
          float q = __builtin_rintf((xv - mn) / st) * st + mn;
          r[e] = (mn == mx) ? xv : q;
        }
        v4f ov;
        ov.x = r[0]; ov.y = r[1]; ov.z = r[2]; ov.w = r[3];
        // output never re-read: non-temporal store keeps L2 clean
        __builtin_nontemporal_store(ov, &o4[i]);
      }
      __syncthreads();  // buffer reuse fence
    }
  }

  // tail (n % TILE) -- zero for this shape, handled generically by block 0
  uint32_t tail = ntiles * TILE;
  if (blockIdx.x == 0) {
    for (uint32_t i = tail + threadIdx.x; i < n; i += blockDim.x) {
      float xv = x[i];
      int ch = 0;
#pragma unroll
      for (int k = 0; k < 15; ++k) ch += (xv >= thr[k]) ? 1 : 0;
      float mn = mnS[ch], mx = mxS[ch], st = stS[ch];
      float q = __builtin_rintf((xv - mn) / st) * st + mn;
      out[i] = (mn == mx) ? xv : q;
    }
  }
}

// ---------------------------------------------------------------------------
extern "C" void kernel_launch(void* const* d_in, const int* in_sizes, int n_in,
                              void* d_out, int out_size, void* d_ws, size_t ws_size,
                              hipStream_t stream) {
  (void)n_in; (void)out_size; (void)ws_size;
  const float* x = (const float*)d_in[0];
  float* out = (float*)d_out;
  uint32_t n = (uint32_t)in_sizes[0];

  uint32_t* w = (uint32_t*)d_ws;
  uint32_t* histHi = w;                                  // 65536 words
  uint32_t* bounds = w + HBINS;                          // 64 words
  uint32_t* histLo = w + HBINS + 64;                     // 32*65536 words (8MB)
  float* vals = (float*)(w + HBINS + 64 + 32 * HBINS);   // 32 floats
  float* tabs = vals + 32;                               // 64 floats

  uint32_t initWords = HBINS + 64 + 32 * HBINS;
  k_init<<<1024, 256, 0, stream>>>(w, initWords);
  k_hist_hi<<<2048, 256, 0, stream>>>(x, n, histHi);
  k_select_hi<<<1, 1024, 0, stream>>>(histHi, n, bounds);
  k_hist_lo<<<2048, 256, 0, stream>>>(x, n, bounds, histLo);
  k_select_lo<<<32, 256, 0, stream>>>(bounds, histLo, vals);
  k_finalize<<<1, 32, 0, stream>>>(vals, tabs);
  k_quant<<<2048, 256, 0, stream>>>(x, out, n, tabs);
}